// SelfAttention_45758581571698
// MI455X (gfx1250) — compile-verified
//
#include <hip/hip_runtime.h>
#include <math.h>

// Problem constants (match reference)
#define BB 2
#define SS 512
#define HH 256
#define AA 128

typedef float v2f __attribute__((ext_vector_type(2)));
typedef float v8f __attribute__((ext_vector_type(8)));

// GEMM macro-tile config
#define BM 32      // rows per workgroup tile
#define BN 64      // cols per workgroup tile
#define KB 16      // K-block staged per buffer
#define NT 128     // threads per workgroup (4 waves, 2x2 wave grid)

// ---------------------------------------------------------------------------
// CDNA5 async global->LDS DMA (ASYNCcnt-tracked), per-lane b128.
//   LDS address comes from the vdst VGPR; global address from a 64-bit VGPR pair
// ---------------------------------------------------------------------------
__device__ __forceinline__ void async_copy_b128(uint32_t lds_off, const float* g)
{
    asm volatile("global_load_async_to_lds_b128 %0, %1, off"
                 :: "v"(lds_off), "v"((unsigned long long)(uintptr_t)g)
                 : "memory");
}

__device__ __forceinline__ void wait_async0()
{
    asm volatile("s_wait_asynccnt 0x0" ::: "memory");
}

// ---------------------------------------------------------------------------
// fp32 WMMA GEMM: C = A(MxK, lda) * B(KxN, ldb), row-major, batched.
// Workgroup = 4 waves, 32x64 macro-tile; each wave owns a 16x32 sub-tile
// (two 16x16 V_WMMA_F32_16X16X4_F32 accumulator chains).
// A/B K-blocks double-buffered in LDS via global_load_async_to_lds_b128.
//
// Fragment layout (wave32, CDNA5 ISA 7.12.2):
//   A (16x4 f32):  lane L, vgpr v -> A[M = L%16][K = v + 2*(L/16)]
//   B (4x16 f32):  lane L, vgpr v -> B[K = v + 2*(L/16)][N = L%16]
//   C/D (16x16):   lane L, vgpr v -> C[M = v + 8*(L/16)][N = L%16]
// ---------------------------------------------------------------------------
__global__ __launch_bounds__(NT) void gemm_f32_wmma(
    const float* __restrict__ Am, const float* __restrict__ Bm,
    float* __restrict__ Cm,
    int K, int lda, int ldb, int ldc,
    long strideA, long strideB, long strideC)
{
    // [A buf0][A buf1][B buf0][B buf1], floats
    __shared__ __align__(16) float smem[2 * (BM * KB) + 2 * (KB * BN)];
    const uint32_t smemBase = (uint32_t)(uintptr_t)&smem[0];

    const int tid  = threadIdx.x;
    const int lane = tid & 31;
    const int w    = tid >> 5;          // wave id 0..3
    const int hh   = lane >> 4;         // lane half
    const int l16  = lane & 15;
    const int wm   = (w >> 1) * 16;     // wave M offset in macro-tile
    const int wn   = (w & 1) * 32;      // wave N offset in macro-tile

    const int n0 = blockIdx.x * BN;
    const int m0 = blockIdx.y * BM;
    const int b  = blockIdx.z;

    const float* A = Am + (long)b * strideA;
    const float* B = Bm + (long)b * strideB;
    float*       C = Cm + (long)b * strideC;

    // --- staging maps (each thread: 1x b128 of A, 2x b128 of B) ---
    const int arow = tid >> 2;                 // 0..31
    const int acol = (tid & 3) * 4;            // 0,4,8,12
    const float* gA = A + (long)(m0 + arow) * lda + acol;
    const uint32_t ldsA = smemBase + (uint32_t)((arow * KB + acol) * 4);

    const int brow = tid >> 3;                 // 0..15
    const int bcol = (tid & 7) * 8;            // 0..56
    const float* gB = B + (long)brow * ldb + n0 + bcol;
    const uint32_t ldsB = smemBase + (uint32_t)((2 * BM * KB + brow * BN + bcol) * 4);

    const uint32_t bufA = (uint32_t)(BM * KB * 4);   // bytes between A buffers
    const uint32_t bufB = (uint32_t)(KB * BN * 4);   // bytes between B buffers

    // prologue: stage K-block 0 into buffer 0
    async_copy_b128(ldsA, gA);
    async_copy_b128(ldsB, gB);
    async_copy_b128(ldsB + 16, gB + 4);

    v8f acc0 = {};
    v8f acc1 = {};

    const int NBK = K / KB;
    int buf = 0;
    for (int blk = 0; blk < NBK; ++blk) {
        wait_async0();        // my DMA done
        __syncthreads();      // everyone's DMA done; prev compute on buf^1 done

        if (blk + 1 < NBK) {  // prefetch next K-block into the other buffer
            const int nb = buf ^ 1;
            const float* gA2 = gA + (blk + 1) * KB;
            const float* gB2 = gB + (long)(blk + 1) * KB * ldb;
            async_copy_b128(ldsA + nb * bufA, gA2);
            async_copy_b128(ldsB + nb * bufB, gB2);
            async_copy_b128(ldsB + nb * bufB + 16, gB2 + 4);
        }

        // compute on current buffer
        const float* sA = &smem[buf * (BM * KB) + (wm + l16) * KB];
        const float* sB = &smem[2 * (BM * KB) + buf * (KB * BN) + wn + l16];
#pragma unroll
        for (int kk = 0; kk < KB; kk += 4) {
            v2f aF = *reinterpret_cast<const v2f*>(&sA[kk + 2 * hh]);
            v2f b0, b1;
            b0.x = sB[(kk + 2 * hh) * BN];
            b0.y = sB[(kk + 2 * hh + 1) * BN];
            b1.x = sB[(kk + 2 * hh) * BN + 16];
            b1.y = sB[(kk + 2 * hh + 1) * BN + 16];
            acc0 = __builtin_amdgcn_wmma_f32_16x16x4_f32(
                false, aF, false, b0, (short)0, acc0, false, false);
            acc1 = __builtin_amdgcn_wmma_f32_16x16x4_f32(
                false, aF, false, b1, (short)0, acc1, false, false);
        }
        buf ^= 1;
    }

#pragma unroll
    for (int v = 0; v < 8; ++v) {
        const long row = (long)(m0 + wm + v + 8 * hh) * ldc;
        C[row + n0 + wn + l16]      = acc0[v];
        C[row + n0 + wn + 16 + l16] = acc1[v];
    }
}

// ---------------------------------------------------------------------------
// s[row] = sum_a query[a] * tanh(k[row][a]);  A = 128 lanes per block
// ---------------------------------------------------------------------------
__global__ __launch_bounds__(AA) void score_kernel(
    const float* __restrict__ kmat, const float* __restrict__ query,
    float* __restrict__ s)
{
    const int row = blockIdx.x;
    const int a   = threadIdx.x;
    __shared__ float red[AA];
    red[a] = query[a] * tanhf(kmat[(long)row * AA + a]);
    __syncthreads();
#pragma unroll
    for (int off = AA / 2; off > 0; off >>= 1) {
        if (a < off) red[a] += red[a + off];
        __syncthreads();
    }
    if (a == 0) s[row] = red[0];
}

// ---------------------------------------------------------------------------
// Prefix softmax statistics per batch (running max m_i, running denom l_i).
// ---------------------------------------------------------------------------
__global__ void prefix_stats_kernel(const float* __restrict__ s,
                                    float* __restrict__ m,
                                    float* __restrict__ l)
{
    const int b = blockIdx.x * blockDim.x + threadIdx.x;
    if (b >= BB) return;
    float mm = -INFINITY, ll = 0.0f;
    for (int i = 0; i < SS; ++i) {
        const float si   = s[b * SS + i];
        const float mnew = fmaxf(mm, si);
        ll = ll * expf(mm - mnew) + expf(si - mnew);
        mm = mnew;
        m[b * SS + i] = mm;
        l[b * SS + i] = ll;
    }
}

// ---------------------------------------------------------------------------
// d[b,i,j] = (j<=i && am[b,i] && am[b,j]) ? exp(s[b,j]-m[b,i]) / l[b,i] : 0
// ---------------------------------------------------------------------------
__global__ __launch_bounds__(256) void fill_d_kernel(
    const float* __restrict__ s, const float* __restrict__ m,
    const float* __restrict__ l, const int* __restrict__ am,
    float* __restrict__ dmat)
{
    const int idx = blockIdx.x * blockDim.x + threadIdx.x;
    const int total = BB * SS * SS;
    if (idx >= total) return;
    const int j = idx % SS;
    const int t = idx / SS;
    const int i = t % SS;
    const int b = t / SS;
    float val = 0.0f;
    if (j <= i && am[b * SS + i] != 0 && am[b * SS + j] != 0) {
        val = expf(s[b * SS + j] - m[b * SS + i]) / l[b * SS + i];
    }
    dmat[idx] = val;
}

// ---------------------------------------------------------------------------
extern "C" void kernel_launch(void* const* d_in, const int* in_sizes, int n_in,
                              void* d_out, int out_size, void* d_ws, size_t ws_size,
                              hipStream_t stream)
{
    (void)in_sizes; (void)n_in; (void)out_size; (void)ws_size;

    const float* x     = (const float*)d_in[0];  // (B,S,H)
    const int*   am    = (const int*)  d_in[1];  // (B,S)
    const float* w_a   = (const float*)d_in[2];  // (H,A)
    const float* query = (const float*)d_in[3];  // (A,)

    float* out   = (float*)d_out;
    float* out_a = out;                              // (B,S,H) first
    float* out_d = out + (long)BB * SS * HH;         // (B,S,S) second

    float* ws   = (float*)d_ws;
    float* kmat = ws;                                // B*S*A
    float* svec = kmat + (long)BB * SS * AA;         // B*S
    float* mvec = svec + BB * SS;                    // B*S
    float* lvec = mvec + BB * SS;                    // B*S

    // 1) k = x @ w_a   (M = B*S = 1024, K = H = 256, N = A = 128)
    {
        dim3 grid(AA / BN, (BB * SS) / BM, 1);
        gemm_f32_wmma<<<grid, NT, 0, stream>>>(
            x, w_a, kmat, HH, HH, AA, AA, 0L, 0L, 0L);
    }

    // 2) s[b,j] = query . tanh(k[b,j,:])
    score_kernel<<<BB * SS, AA, 0, stream>>>(kmat, query, svec);

    // 3) prefix softmax stats per batch
    prefix_stats_kernel<<<1, 32, 0, stream>>>(svec, mvec, lvec);

    // 4) materialize d (written directly into its d_out slice)
    {
        const int total = BB * SS * SS;
        fill_d_kernel<<<(total + 255) / 256, 256, 0, stream>>>(
            svec, mvec, lvec, am, out_d);
    }

    // 5) a = d @ x per batch  (M = S = 512, K = S = 512, N = H = 256)
    {
        dim3 grid(HH / BN, SS / BM, BB);
        gemm_f32_wmma<<<grid, NT, 0, stream>>>(
            out_d, x, out_a, SS, SS, HH, HH,
            (long)SS * SS, (long)SS * HH, (long)SS * HH);
    }
}